// reservoir_2473901162771
// MI455X (gfx1250) — compile-verified
//
#include <hip/hip_runtime.h>

typedef __attribute__((ext_vector_type(2))) float v2f;
typedef __attribute__((ext_vector_type(8))) float v8f;

#define RESV 4096
#define INPD 512
#define SEQT 4096
#define SCAN_WGS 256          // 16 rows of W_hat per WG
#define SCAN_THREADS 256      // 8 waves -> 2 rows per wave

// ---------------------------------------------------------------------------
// Kernel A: U = X @ Win^T  via V_WMMA_F32_16X16X4_F32
// X:   (SEQT, INPD) row-major      (M x K)
// Win: (RESV, INPD) row-major  ->  B[k][n] = Win[n][k]   (K x N)
// U:   (SEQT, RESV) row-major, written into d_out
// Each wave computes one 16x16 tile; block = 8 waves = 128 rows x 16 cols.
// ---------------------------------------------------------------------------
__global__ __launch_bounds__(256) void u_gemm_wmma(const float* __restrict__ X,
                                                   const float* __restrict__ Win,
                                                   float* __restrict__ U) {
    const int K = INPD, N = RESV;
    const int wave = threadIdx.x >> 5;
    const int lane = threadIdx.x & 31;
    const int hi   = lane >> 4;     // 0: K=0,1  1: K=2,3 within each 4-step
    const int l    = lane & 15;

    const int n0 = blockIdx.x * 16;
    const int m0 = (blockIdx.y * 8 + wave) * 16;

    // A fragment source: lane (hi,l) holds X[m0+l][k + 2*hi .. +1]
    const float* arow = X   + (size_t)(m0 + l) * K + 2 * hi;
    // B fragment source: lane (hi,l) holds Win[n0+l][k + 2*hi .. +1] = B[k+2*hi..][n0+l]
    const float* brow = Win + (size_t)(n0 + l) * K + 2 * hi;

    v8f c = {};
    for (int k = 0; k < K; k += 4) {
        v2f a = *(const v2f*)(arow + k);
        v2f b = *(const v2f*)(brow + k);
        c = __builtin_amdgcn_wmma_f32_16x16x4_f32(
                /*neg_a=*/false, a, /*neg_b=*/false, b,
                /*c_mod=*/(short)0, c, /*reuse_a=*/false, /*reuse_b=*/false);
    }

    // C/D layout: VGPR r -> lanes 0-15: (M=m0+r, N=n0+l); lanes 16-31: (M=m0+8+r, N=n0+l)
    float* out = U + (size_t)(m0 + hi * 8) * N + n0 + l;
#pragma unroll
    for (int r = 0; r < 8; ++r) out[(size_t)r * N] = c[r];
}

// ---------------------------------------------------------------------------
// Barrier state init (d_ws is poisoned 0xAA by the harness; re-zero each call)
// ---------------------------------------------------------------------------
__global__ void init_barrier(unsigned* bar) {
    bar[0] = 0u;   // arrival count
    bar[1] = 0u;   // generation
}

// ---------------------------------------------------------------------------
// Kernel B: persistent sequential scan.
//   h_t = relu(U_t + W_hat @ h_{t-1}),  U_t lives in out[t] and is overwritten.
// 256 WGs x 256 threads; WG b owns rows [16b, 16b+16); wave w owns 2 rows.
// Device-wide generation barrier between steps (count/gen in d_ws).
// ---------------------------------------------------------------------------
__global__ __launch_bounds__(SCAN_THREADS) void scan_kernel(
        const float* __restrict__ What,   // (RESV, RESV)
        const float* __restrict__ h0,     // (RESV)
        float* __restrict__ out,          // (SEQT, RESV): U in, states out
        unsigned* bar_count, unsigned* bar_gen) {
    __shared__ __align__(16) float hs[RESV];

    const int tid  = threadIdx.x;
    const int wave = tid >> 5;
    const int lane = tid & 31;
    const int row0 = blockIdx.x * 16 + wave * 2;

    for (int t = 0; t < SEQT; ++t) {
        // stage h_{t-1} into LDS (16 KB), float4-vectorized
        const float* hsrc = (t == 0) ? h0 : (out + (size_t)(t - 1) * RESV);
        for (int i = tid * 4; i < RESV; i += SCAN_THREADS * 4) {
            *reinterpret_cast<float4*>(hs + i) =
                *reinterpret_cast<const float4*>(hsrc + i);
        }
        __syncthreads();

#pragma unroll
        for (int rr = 0; rr < 2; ++rr) {
            const int row = row0 + rr;
            const float* wrow = What + (size_t)row * RESV;
            float acc = 0.0f;
            // lane-parallel float4 dot: 32 iterations, 16B per lane per iter
            for (int j = lane * 4; j < RESV; j += 32 * 4) {
                float4 w  = *reinterpret_cast<const float4*>(wrow + j);
                float4 h4 = *reinterpret_cast<const float4*>(hs + j);
                acc = fmaf(w.x, h4.x, acc);
                acc = fmaf(w.y, h4.y, acc);
                acc = fmaf(w.z, h4.z, acc);
                acc = fmaf(w.w, h4.w, acc);
            }
#pragma unroll
            for (int off = 16; off > 0; off >>= 1)
                acc += __shfl_xor(acc, off, 32);
            if (lane == 0) {
                float* o = out + (size_t)t * RESV + row;
                float u = *o;                       // U_t[row]
                *o = fmaxf(u + acc, 0.0f);          // h_t[row]
            }
        }

        // ---- device-wide barrier (sense-free, monotonic generation) ----
        __threadfence();        // publish h_t rows
        __syncthreads();        // whole WG done with step t
        if (tid == 0) {
            const unsigned target = (unsigned)(t + 1);
            unsigned arrived = atomicAdd(bar_count, 1u);
            if (arrived == SCAN_WGS - 1u) {
                atomicExch(bar_count, 0u);   // reset for next round...
                __threadfence();             // ...visible before release
                atomicExch(bar_gen, target); // release
            } else {
                while (atomicAdd(bar_gen, 0u) < target) {
                    __builtin_amdgcn_s_sleep(2);
                }
            }
        }
        __syncthreads();
        __threadfence();        // acquire: see other WGs' h_t before reload
    }
}

// ---------------------------------------------------------------------------
extern "C" void kernel_launch(void* const* d_in, const int* in_sizes, int n_in,
                              void* d_out, int out_size, void* d_ws, size_t ws_size,
                              hipStream_t stream) {
    const float* X    = (const float*)d_in[0];  // (SEQT, INPD)
    const float* Win  = (const float*)d_in[1];  // (RESV, INPD)
    const float* What = (const float*)d_in[2];  // (RESV, RESV)
    const float* h0   = (const float*)d_in[3];  // (RESV)
    float* out        = (float*)d_out;          // (SEQT, RESV)
    unsigned* bar     = (unsigned*)d_ws;

    // Phase 1: U = X @ Win^T into d_out (WMMA f32)
    dim3 gridA(RESV / 16, SEQT / 128);
    u_gemm_wmma<<<gridA, 256, 0, stream>>>(X, Win, out);

    // Phase 2: persistent scan with device-wide barrier
    init_barrier<<<1, 1, 0, stream>>>(bar);
    scan_kernel<<<SCAN_WGS, SCAN_THREADS, 0, stream>>>(What, h0, out,
                                                       bar, bar + 1);
}